// BankedLinear_22531398435543
// MI455X (gfx1250) — compile-verified
//
#include <hip/hip_runtime.h>

typedef __attribute__((ext_vector_type(2))) float v2f;
typedef __attribute__((ext_vector_type(8))) float v8f;

#define NB   16
#define ROWS 1024   // B*S*K = 2*256*2
#define IN   512
#define OUT  512

// ---------------------------------------------------------------------------
// Kernel 1: bin the 1024 rows by selected bank.
// One workgroup of 1024 threads; LDS atomics give each row a slot in its
// bank's list. Order within a bank is nondeterministic, but every output
// element's value is order-independent (full-K reduction per row), so the
// final d_out is deterministic.
// ws layout (ints): counts[16] | rows[16][1024]
// ---------------------------------------------------------------------------
__global__ __launch_bounds__(ROWS) void BankedLinear_bin_kernel(
    const int* __restrict__ sel, int* __restrict__ counts,
    int* __restrict__ rows) {
  __shared__ int lcnt[NB];
  const int t = threadIdx.x;
  if (t < NB) lcnt[t] = 0;
  __syncthreads();
  const int bank = sel[t] & (NB - 1);
  const int pos = atomicAdd(&lcnt[bank], 1);
  rows[bank * ROWS + pos] = t;
  __syncthreads();
  if (t < NB) counts[t] = lcnt[t];
}

// ---------------------------------------------------------------------------
// Kernel 2: per-bank GEMM with V_WMMA_F32_16X16X4_F32.
// grid = (OUT/64, 64 max M-tiles, NB banks); block = 128 threads (4 waves).
// Each wave computes one 16(M) x 16(N) f32 tile, reducing over K = IN = 512
// in steps of 4 (two interleaved accumulators to break the WMMA D->C chain).
//
// A 16x4 f32 layout (ISA 7.12.2): lanes 0-15 hold {K=0,K=1} for M=lane,
// lanes 16-31 hold {K=2,K=3}  -> per-lane float2 load at xrow + k0 + 2*(lane>>4).
// B 4x16 f32 layout (symmetric): lane n<16 holds {K=0,K=1} for N=n, lanes
// 16-31 hold {K=2,K=3}         -> per-lane float2 load from weight row N.
// C/D: VGPR r, lanes 0-15 -> (M=r, N=lane); lanes 16-31 -> (M=r+8, N=lane-16).
// ---------------------------------------------------------------------------
__global__ __launch_bounds__(128) void BankedLinear_gemm_kernel(
    const float* __restrict__ x, const float* __restrict__ w,
    const float* __restrict__ bias, const int* __restrict__ counts,
    const int* __restrict__ rows, float* __restrict__ out) {
  const int bank = blockIdx.z;
  const int cnt  = counts[bank];
  const int mt   = blockIdx.y;
  if (mt * 16 >= cnt) return;            // uniform exit: EXEC stays all-1s

  const int lane  = threadIdx.x & 31;
  const int wave  = threadIdx.x >> 5;
  const int n_base = (blockIdx.x * 4 + wave) * 16;
  const int nlane = lane & 15;           // N column / A row within tile
  const int half  = lane >> 4;           // 0: K=0,1 ; 1: K=2,3
  const int koff  = half * 2;

  const int* tile_rows = rows + bank * ROWS + mt * 16;
  const int mslot = mt * 16 + nlane;
  const int arow  = (mslot < cnt) ? tile_rows[nlane] : tile_rows[0];

  const float* xrow = x + (size_t)arow * IN + koff;
  const float* wrow = w + ((size_t)bank * OUT + n_base + nlane) * IN + koff;

  v8f acc0 = {0.f, 0.f, 0.f, 0.f, 0.f, 0.f, 0.f, 0.f};
  v8f acc1 = acc0;

#pragma unroll 4
  for (int k0 = 0; k0 < IN; k0 += 8) {
    const v2f a0 = *(const v2f*)(xrow + k0);
    const v2f b0 = *(const v2f*)(wrow + k0);
    const v2f a1 = *(const v2f*)(xrow + k0 + 4);
    const v2f b1 = *(const v2f*)(wrow + k0 + 4);
    // 8 args: (neg_a, A, neg_b, B, c_mod, C, reuse_a, reuse_b)
    acc0 = __builtin_amdgcn_wmma_f32_16x16x4_f32(
        false, a0, false, b0, (short)0, acc0, false, false);
    acc1 = __builtin_amdgcn_wmma_f32_16x16x4_f32(
        false, a1, false, b1, (short)0, acc1, false, false);
  }
  v8f acc = acc0 + acc1;

  const float bv = bias[bank * OUT + n_base + nlane];  // N-only, fixed per lane

#pragma unroll
  for (int r = 0; r < 8; ++r) {
    const int ms = mt * 16 + r + 8 * half;             // M slot of acc[r]
    if (ms < cnt) {
      const int orow = rows[bank * ROWS + ms];
      out[(size_t)orow * OUT + n_base + nlane] = acc[r] + bv;
    }
  }
}

extern "C" void kernel_launch(void* const* d_in, const int* in_sizes, int n_in,
                              void* d_out, int out_size, void* d_ws,
                              size_t ws_size, hipStream_t stream) {
  (void)in_sizes; (void)n_in; (void)out_size; (void)ws_size;
  const float* x    = (const float*)d_in[0];  // (B,S,K,IN) f32
  const int*   sel  = (const int*)d_in[1];    // (B,S,K)    i32
  const float* w    = (const float*)d_in[2];  // (NB,OUT,IN) f32
  const float* bias = (const float*)d_in[3];  // (NB,OUT)   f32
  float* out = (float*)d_out;                 // (B,S,K,OUT) f32

  int* counts = (int*)d_ws;        // 16 ints
  int* rows   = counts + NB;       // 16*1024 ints

  BankedLinear_bin_kernel<<<1, ROWS, 0, stream>>>(sel, counts, rows);

  dim3 grid(OUT / 64, ROWS / 16, NB);   // (8 N-tiles, 64 max M-tiles, 16 banks)
  BankedLinear_gemm_kernel<<<grid, 128, 0, stream>>>(x, w, bias, counts, rows,
                                                     out);
}